// Model_65335042507147
// MI455X (gfx1250) — compile-verified
//
#include <hip/hip_runtime.h>

// Problem constants (match reference setup_inputs()).
#define G_DIM          2
#define NUM_REQS       1024
#define MAX_NUM_REQS   4096
#define MAX_NUM_BLOCKS 8192

#define THREADS 256
#define ITERS   (MAX_NUM_BLOCKS / 4 / THREADS)   // 8192 elems / 4 per chunk / 256 threads = 8

// ---- CDNA5 async global<->LDS path (guarded: compile-only environment) ----
#if defined(__HIP_DEVICE_COMPILE__) && defined(__gfx1250__) && \
    __has_builtin(__builtin_amdgcn_global_load_async_to_lds_b128) && \
    __has_builtin(__builtin_amdgcn_global_store_async_from_lds_b128)
#define USE_ASYNC 1
#else
#define USE_ASYNC 0
#endif

#if USE_ASYNC
// Builtin prototype (from clang diagnostic): first two params are pointers to
// 'int __attribute__((vector_size(16)))' in addrspace(1) / addrspace(3).
typedef int v4i __attribute__((vector_size(4 * sizeof(int))));
typedef __attribute__((address_space(1))) v4i* g_v4i_ptr;   // global (prints as __device__)
typedef __attribute__((address_space(3))) v4i* l_v4i_ptr;   // LDS    (prints as __shared__)

// Global generic addresses map 1:1 onto addrspace(1).
__device__ __forceinline__ g_v4i_ptr to_global(const void* p) {
    return (g_v4i_ptr)(unsigned long long)p;
}
// Generic pointers to LDS carry the LDS byte offset in their low 32 bits
// (flat-aperture rule); standard CK/aiter idiom.
__device__ __forceinline__ l_v4i_ptr to_lds(const void* p) {
    return (l_v4i_ptr)(unsigned int)(unsigned long long)p;
}

__device__ __forceinline__ void wait_async0() {
#if __has_builtin(__builtin_amdgcn_s_wait_asynccnt)
    __builtin_amdgcn_s_wait_asynccnt(0);
#else
    asm volatile("s_wait_asynccnt 0" ::: "memory");
#endif
}
#endif // USE_ASYNC

__global__ __launch_bounds__(THREADS)
void block_table_gather_kernel(const int* __restrict__ idx_mapping,   // [NUM_REQS]
                               const int* __restrict__ src,           // [G, MAX_NUM_REQS, MAX_NUM_BLOCKS]
                               const int* __restrict__ num_blocks,    // [G, MAX_NUM_REQS]
                               const int* __restrict__ dst_in,        // [G, NUM_REQS, MAX_NUM_BLOCKS]
                               int*       __restrict__ out)           // [G, NUM_REQS, MAX_NUM_BLOCKS]
{
    const int r   = blockIdx.x;       // request row
    const int g   = blockIdx.y;       // group
    const int tid = threadIdx.x;

    // Uniform per-block scalars (compiler scalarizes: blockIdx-only addressing).
    const int idx = idx_mapping[r];
    const int n   = num_blocks[(size_t)g * MAX_NUM_REQS + idx];

    const int* srcRow = src    + ((size_t)g * MAX_NUM_REQS + idx) * MAX_NUM_BLOCKS;
    const int* dstRow = dst_in + ((size_t)g * NUM_REQS     + r  ) * MAX_NUM_BLOCKS;
    int*       outRow = out    + ((size_t)g * NUM_REQS     + r  ) * MAX_NUM_BLOCKS;

#if USE_ASYNC
    __shared__ int row[MAX_NUM_BLOCKS];   // 32 KB staging buffer for one row

    // Phase 1: DMA the selected row into LDS with async B128 loads.
    #pragma unroll
    for (int i = 0; i < ITERS; ++i) {
        const int c = (i * THREADS + tid) * 4;       // 16B-aligned chunk start
        if (c + 4 <= n) {
            __builtin_amdgcn_global_load_async_to_lds_b128(
                to_global(srcRow + c), to_lds(&row[c]), 0, 0);
        } else if (c >= n) {
            __builtin_amdgcn_global_load_async_to_lds_b128(
                to_global(dstRow + c), to_lds(&row[c]), 0, 0);
        } else {
            // The single chunk straddling n: resolve element-wise.
            int4 v;
            v.x = (c + 0 < n) ? srcRow[c + 0] : dstRow[c + 0];
            v.y = (c + 1 < n) ? srcRow[c + 1] : dstRow[c + 1];
            v.z = (c + 2 < n) ? srcRow[c + 2] : dstRow[c + 2];
            v.w = (c + 3 < n) ? srcRow[c + 3] : dstRow[c + 3];
            *reinterpret_cast<int4*>(&row[c]) = v;   // ds_store_b128
        }
    }
    wait_async0();        // all async loads landed in LDS
    __syncthreads();      // plus DScnt drain for the straddle chunk

    // Phase 2: DMA the assembled row from LDS to the output with async B128 stores.
    #pragma unroll
    for (int i = 0; i < ITERS; ++i) {
        const int c = (i * THREADS + tid) * 4;
        __builtin_amdgcn_global_store_async_from_lds_b128(
            to_global(outRow + c), to_lds(&row[c]), 0, 0);
    }
    wait_async0();        // drain before wave end (S_ENDPGM wait-idle backstop)
#else
    // Fallback: direct coalesced int4 copy (bandwidth-optimal baseline).
    #pragma unroll
    for (int i = 0; i < ITERS; ++i) {
        const int c = (i * THREADS + tid) * 4;
        int4 v;
        if (c + 4 <= n) {
            v = *reinterpret_cast<const int4*>(srcRow + c);
        } else if (c >= n) {
            v = *reinterpret_cast<const int4*>(dstRow + c);
        } else {
            v.x = (c + 0 < n) ? srcRow[c + 0] : dstRow[c + 0];
            v.y = (c + 1 < n) ? srcRow[c + 1] : dstRow[c + 1];
            v.z = (c + 2 < n) ? srcRow[c + 2] : dstRow[c + 2];
            v.w = (c + 3 < n) ? srcRow[c + 3] : dstRow[c + 3];
        }
        *reinterpret_cast<int4*>(outRow + c) = v;
    }
#endif
}

extern "C" void kernel_launch(void* const* d_in, const int* in_sizes, int n_in,
                              void* d_out, int out_size, void* d_ws, size_t ws_size,
                              hipStream_t stream) {
    // setup_inputs() order: idx_mapping, src_block_tables, num_blocks, dst_block_tables
    const int* idx_mapping = (const int*)d_in[0];
    const int* src         = (const int*)d_in[1];
    const int* num_blocks  = (const int*)d_in[2];
    const int* dst_in      = (const int*)d_in[3];
    int*       out         = (int*)d_out;   // int32 output, same 4-byte width

    dim3 grid(NUM_REQS, G_DIM);             // one workgroup per output row
    block_table_gather_kernel<<<grid, THREADS, 0, stream>>>(
        idx_mapping, src, num_blocks, dst_in, out);
}